// MovieRecommenderEngine_2087354105996
// MI455X (gfx1250) — compile-verified
//
#include <hip/hip_runtime.h>

// GraphSAGE 2-layer forward for MI455X (gfx1250, wave32).
// Edge aggregation: L2-resident f32 atomic scatter (x/h/sums all fit in 192MB L2).
// GEMMs: fp32 WMMA (V_WMMA_F32_16X16X4_F32), mean-division fused into the
// A-fragment load, lin_l/lin_r fused as two branch-free K-halves.
// Compile-time KH/N -> strength-reduced addressing; 2 accumulators per wave
// for WMMA ILP; unroll-8 so loads clause ahead of the WMMA group.

#define NODES 40000
#define IN_C  128
#define HID_C 256
#define OUT_C 128

typedef __attribute__((ext_vector_type(2))) float v2f;
typedef __attribute__((ext_vector_type(8))) float v8f;

// ---------------- degree / inverse-count ----------------
__global__ void sage_degree_kernel(const int* __restrict__ dst,
                                   float* __restrict__ cnt, int E) {
    int e = blockIdx.x * blockDim.x + threadIdx.x;
    if (e < E) atomicAdd(&cnt[dst[e]], 1.0f);
}

__global__ void sage_invcnt_kernel(const float* __restrict__ cnt,
                                   float* __restrict__ invc, int n) {
    int i = blockIdx.x * blockDim.x + threadIdx.x;
    if (i < n) invc[i] = 1.0f / fmaxf(cnt[i], 1.0f);
}

// ---------------- edge scatter-add (one wave32 per edge) ----------------
template <int F>
__global__ void sage_scatter_kernel(const int* __restrict__ src,
                                    const int* __restrict__ dst,
                                    const float* __restrict__ feat,
                                    float* __restrict__ sums, int E) {
    int w    = (blockIdx.x * blockDim.x + threadIdx.x) >> 5;  // edge id
    int lane = threadIdx.x & 31;
    if (w >= E) return;
    int s = src[w], d = dst[w];
    const float* fr = feat + (size_t)s * F;
    float*       so = sums + (size_t)d * F;
#pragma unroll
    for (int j = 0; j < F / 32; ++j) {
        int f = lane + 32 * j;                 // coalesced across the wave
        atomicAdd(&so[f], fr[f]);              // global_atomic_add_f32 -> L2
    }
}

// ---------------- fused SAGE GEMM: out = (S*inv) @ Wl + X @ Wr + b ----------
// Wave computes a 16(M) x 32(N) strip: one A fragment, two B fragments, two
// independent accumulator chains. 8 waves/block -> 64(M) x 64(N) block tile.
// fp32 WMMA fragment layouts (ISA 7.12.2):
//   A 16x4 : lane L holds row m=L&15, K = (L>>4)*2 + {0,1}
//   B 4x16 : lane L holds col n=L&15, K = (L>>4)*2 + {0,1}
//   C 16x16: VGPR v -> row (L>>4)*8 + v, col L&15
template <int KH, int N, int RELU>
__global__ __launch_bounds__(256)
void sage_gemm(const float* __restrict__ S,     // [M,KH] neighbor sums
               const float* __restrict__ X,     // [M,KH] self features
               const float* __restrict__ invc,  // [M] 1/max(deg,1)
               const float* __restrict__ Wl,    // [KH,N]
               const float* __restrict__ Wr,    // [KH,N]
               const float* __restrict__ bias,  // [N]
               float* __restrict__ out) {       // [M,N]
    const int lane  = threadIdx.x & 31;
    const int wave  = threadIdx.x >> 5;
    const int g     = lane >> 4;       // half-wave group
    const int l15   = lane & 15;
    const int waveM = wave >> 1;       // 0..3
    const int waveN = wave & 1;        // 0..1

    const int row  = blockIdx.y * 64 + waveM * 16 + l15;   // A row this lane
    const int col0 = blockIdx.x * 64 + waveN * 32 + l15;   // B col, tile 0
    const int col1 = col0 + 16;                            // B col, tile 1

    const float inv = invc[row];

    v8f c0 = {}, c1 = {};

    // ---- half 1: mean = S*inv, times Wl ----
    {
        const float* Ap = S  + (size_t)row * KH + g * 2;
        const float* Wp = Wl + (size_t)(g * 2) * N + col0;
#pragma unroll 8
        for (int k0 = 0; k0 < KH; k0 += 4) {
            float2 av = *(const float2*)Ap;
            v2f a;  a[0]  = av.x * inv;  a[1]  = av.y * inv;
            v2f b0; b0[0] = Wp[0];       b0[1] = Wp[N];
            v2f b1; b1[0] = Wp[16];      b1[1] = Wp[N + 16];
            c0 = __builtin_amdgcn_wmma_f32_16x16x4_f32(
                     false, a, false, b0, (short)0, c0, false, false);
            c1 = __builtin_amdgcn_wmma_f32_16x16x4_f32(
                     false, a, false, b1, (short)0, c1, false, false);
            Ap += 4;
            Wp += 4 * N;
        }
    }

    // ---- half 2: X times Wr ----
    {
        const float* Ap = X  + (size_t)row * KH + g * 2;
        const float* Wp = Wr + (size_t)(g * 2) * N + col0;
#pragma unroll 8
        for (int k0 = 0; k0 < KH; k0 += 4) {
            float2 av = *(const float2*)Ap;
            v2f a;  a[0]  = av.x;        a[1]  = av.y;
            v2f b0; b0[0] = Wp[0];       b0[1] = Wp[N];
            v2f b1; b1[0] = Wp[16];      b1[1] = Wp[N + 16];
            c0 = __builtin_amdgcn_wmma_f32_16x16x4_f32(
                     false, a, false, b0, (short)0, c0, false, false);
            c1 = __builtin_amdgcn_wmma_f32_16x16x4_f32(
                     false, a, false, b1, (short)0, c1, false, false);
            Ap += 4;
            Wp += 4 * N;
        }
    }

    const float bv0 = bias[col0];
    const float bv1 = bias[col1];
    float* orow = out + (size_t)(blockIdx.y * 64 + waveM * 16 + g * 8) * N + col0;
#pragma unroll
    for (int v = 0; v < 8; ++v) {
        float v0 = c0[v] + bv0;
        float v1 = c1[v] + bv1;
        if (RELU) { v0 = fmaxf(v0, 0.0f); v1 = fmaxf(v1, 0.0f); }
        orow[(size_t)v * N]      = v0;
        orow[(size_t)v * N + 16] = v1;
    }
}

// ---------------- host launcher ----------------
extern "C" void kernel_launch(void* const* d_in, const int* in_sizes, int n_in,
                              void* d_out, int out_size, void* d_ws, size_t ws_size,
                              hipStream_t stream) {
    const float* x   = (const float*)d_in[0];
    const int*   ei  = (const int*)d_in[1];     // edge_index [2,E] flat (int32)
    const float* W1l = (const float*)d_in[2];
    const float* W1r = (const float*)d_in[3];
    const float* b1  = (const float*)d_in[4];
    const float* W2l = (const float*)d_in[5];
    const float* W2r = (const float*)d_in[6];
    const float* b2  = (const float*)d_in[7];
    float*       out = (float*)d_out;

    const int E = in_sizes[1] / 2;
    const int* srcv = ei;
    const int* dstv = ei + E;

    // Workspace layout (floats). Zeroed region first so one memset covers it.
    float* ws   = (float*)d_ws;
    float* cnt  = ws;                              // NODES
    float* sum1 = cnt  + NODES;                    // NODES*IN_C
    float* sum2 = sum1 + (size_t)NODES * IN_C;     // NODES*HID_C
    float* invc = sum2 + (size_t)NODES * HID_C;    // NODES
    float* h    = invc + NODES;                    // NODES*HID_C

    size_t zero_bytes =
        (size_t)(NODES + (size_t)NODES * IN_C + (size_t)NODES * HID_C) * sizeof(float);
    hipMemsetAsync(d_ws, 0, zero_bytes, stream);

    // degrees + inverse counts (shared by both layers)
    sage_degree_kernel<<<(E + 255) / 256, 256, 0, stream>>>(dstv, cnt, E);
    sage_invcnt_kernel<<<(NODES + 255) / 256, 256, 0, stream>>>(cnt, invc, NODES);

    // layer 1: aggregate x, GEMM (mean@W1_l + x@W1_r + b1), ReLU -> h
    {
        sage_scatter_kernel<IN_C><<<(E * 32 + 255) / 256, 256, 0, stream>>>(
            srcv, dstv, x, sum1, E);
        dim3 grid(HID_C / 64, NODES / 64);
        sage_gemm<IN_C, HID_C, 1><<<grid, 256, 0, stream>>>(
            sum1, x, invc, W1l, W1r, b1, h);
    }

    // layer 2: aggregate h, GEMM (mean@W2_l + h@W2_r + b2) -> out
    {
        sage_scatter_kernel<HID_C><<<(E * 32 + 255) / 256, 256, 0, stream>>>(
            srcv, dstv, h, sum2, E);
        dim3 grid(OUT_C / 64, NODES / 64);
        sage_gemm<HID_C, OUT_C, 0><<<grid, 256, 0, stream>>>(
            sum2, h, invc, W2l, W2r, b2, out);
    }
}